// MyLSTM_38293928411264
// MI455X (gfx1250) — compile-verified
//
#include <hip/hip_runtime.h>
#include <math.h>

// ---------------------------------------------------------------------------
// LSTM on MI455X (gfx1250, wave32)
//   VOCAB=128, HID=1024, SEQ=16384, batch=1
//
// Design:
//  * lstm_scan: persistent kernel, 128 workgroups x 256 threads, grid barrier
//    per timestep via an L2 atomic counter. Each WG owns 8 hidden units
//    (the 4 gate rows each), W_hh slice streamed from L2 (whole W_hh = 16 MB,
//    L2-resident on the 192 MB L2). Input projection W_ih@x_t fused in.
//    f32 VALU FMA is the right engine for a batch-1 matvec.
//  * lstm_out_gemm: out = hs @ W_out^T + b_out with V_WMMA_F32_16X16X4_F32.
//    Each wave computes a 64x16 strip = 4 independent 16x16 accumulators:
//    B fragment reused 4x per K-step (1.25 loads/WMMA) and the 4 WMMAs per
//    step are hazard-free (distinct D registers).
// ---------------------------------------------------------------------------

#define VOCAB 128
#define HID   1024
#define SEQ   16384

#define NWG   128            // workgroups in the persistent scan kernel
#define TPB   256            // 8 waves per workgroup
#define HPW   (HID / NWG)    // hidden units owned per workgroup = 8

typedef __attribute__((ext_vector_type(2))) float v2f;
typedef __attribute__((ext_vector_type(8))) float v8f;

__device__ __forceinline__ float sigmoidf_fast(float x) {
    return 1.0f / (1.0f + __expf(-x));
}

// ---------------------------------------------------------------------------
// Persistent recurrent scan with device-wide barrier per step.
// Thread mapping (per workgroup): 32 "row tasks" (4 gates x 8 units),
// 8 threads per task; each thread reduces 128 W_hh columns + 16 W_ih columns.
// ---------------------------------------------------------------------------
__global__ __launch_bounds__(TPB) void lstm_scan(
    const float* __restrict__ x,      // (SEQ, VOCAB)
    const float* __restrict__ W_ih,   // (4*HID, VOCAB)
    const float* __restrict__ W_hh,   // (4*HID, HID)
    const float* __restrict__ b_ih,   // (4*HID)
    const float* __restrict__ b_hh,   // (4*HID)
    const float* __restrict__ h0,     // (HID)
    const float* __restrict__ c0,     // (HID)
    float* __restrict__ hs,           // (SEQ, HID) output history
    unsigned* __restrict__ counter)   // grid barrier counter (pre-zeroed)
{
    __shared__ float h_s[HID];
    __shared__ float x_s[VOCAB];
    __shared__ float gate_s[4 * HPW];

    const int tid  = threadIdx.x;
    const int wg   = blockIdx.x;
    const int task = tid >> 3;        // 0..31 : gate*HPW + unit
    const int seg  = tid & 7;         // 0..7  : column segment
    const int unit = task & (HPW - 1);
    const int gate = task >> 3;       // 0..3 (i, f, g, o)
    const int j    = wg * HPW + unit; // owned hidden index
    const int row  = gate * HID + j;  // W row

    const float  bias = b_ih[row] + b_hh[row];
    const float* wrow = W_hh + (size_t)row * HID;
    const float* xrow = W_ih + (size_t)row * VOCAB;

    // cell state for the 8 owned units lives in registers of threads 0..7
    float c_reg = (tid < HPW) ? c0[wg * HPW + tid] : 0.0f;

    for (int t = 0; t < SEQ; ++t) {
        // stage h_{t-1} and x_t into LDS (coalesced float4)
        const float* hsrc = (t == 0) ? h0 : (hs + (size_t)(t - 1) * HID);
        ((float4*)h_s)[tid] = ((const float4*)hsrc)[tid];
        if (tid < VOCAB / 4)
            ((float4*)x_s)[tid] = ((const float4*)(x + (size_t)t * VOCAB))[tid];
        __syncthreads();

        // partial dot product: strided by 8 so 8 lanes of a task coalesce
        float acc = 0.0f;
#pragma unroll 8
        for (int i = 0; i < HID / 8; ++i)
            acc = fmaf(wrow[seg + 8 * i], h_s[seg + 8 * i], acc);
#pragma unroll
        for (int i = 0; i < VOCAB / 8; ++i)
            acc = fmaf(xrow[seg + 8 * i], x_s[seg + 8 * i], acc);

        // reduce the 8-lane group (wave32 shuffles)
        acc += __shfl_xor(acc, 1, 32);
        acc += __shfl_xor(acc, 2, 32);
        acc += __shfl_xor(acc, 4, 32);
        if (seg == 0) gate_s[task] = acc + bias;
        __syncthreads();

        // gate math + state update for the 8 owned units
        if (tid < HPW) {
            const float gi = gate_s[tid];
            const float gf = gate_s[HPW + tid];
            const float gg = gate_s[2 * HPW + tid];
            const float go = gate_s[3 * HPW + tid];
            const float iv = sigmoidf_fast(gi);
            const float fv = sigmoidf_fast(gf);
            const float gv = tanhf(gg);
            const float ov = sigmoidf_fast(go);
            c_reg = fv * c_reg + iv * gv;
            hs[(size_t)t * HID + wg * HPW + tid] = ov * tanhf(c_reg);
        }

        // device-wide barrier: release h_t, wait for all 128 WGs
        __threadfence();
        __syncthreads();
        if (tid == 0) {
            __hip_atomic_fetch_add(counter, 1u, __ATOMIC_ACQ_REL,
                                   __HIP_MEMORY_SCOPE_AGENT);
            const unsigned target = (unsigned)(t + 1) * (unsigned)NWG;
            while (__hip_atomic_load(counter, __ATOMIC_ACQUIRE,
                                     __HIP_MEMORY_SCOPE_AGENT) < target) {
                __builtin_amdgcn_s_sleep(1);
            }
        }
        __syncthreads();
    }
}

// ---------------------------------------------------------------------------
// out = hs @ W_out^T + b_out   via V_WMMA_F32_16X16X4_F32
// Grid: SEQ/64 blocks x 8 waves; each wave owns a 64(rows) x 16(vocab) strip
// = four independent 16x16 accumulator tiles.
// A layout (16x4 f32): lane l<16 -> M=l, K=k+{0,1}; lane>=16 -> M=l-16, K=k+{2,3}
// B layout (4x16 f32): lane l<16 -> K=k+{0,1}, N=l; lane>=16 -> K=k+{2,3}, N=l-16
// C layout: VGPR r, lanes 0-15 -> M=r, lanes 16-31 -> M=r+8.
// ---------------------------------------------------------------------------
__global__ __launch_bounds__(256) void lstm_out_gemm(
    const float* __restrict__ hs,     // (SEQ, HID)
    const float* __restrict__ W_out,  // (VOCAB, HID)
    const float* __restrict__ b_out,  // (VOCAB)
    float* __restrict__ out)          // (SEQ, VOCAB)
{
    const int lane = threadIdx.x & 31;
    const int wave = threadIdx.x >> 5;  // 0..7 -> vocab tile
    const int t0   = blockIdx.x * 64;
    const int n0   = wave * 16;
    const int mn   = lane & 15;
    const int hi   = lane >> 4;         // 0 or 1 -> K sub-pair

    const float* aBase = hs    + (size_t)(t0 + mn) * HID + hi * 2;
    const float* bBase = W_out + (size_t)(n0 + mn) * HID + hi * 2;

    v8f c0v = {}, c1v = {}, c2v = {}, c3v = {};
#pragma unroll 2
    for (int k = 0; k < HID; k += 4) {
        const v2f b  = *(const v2f*)(bBase + k);
        const v2f a0 = *(const v2f*)(aBase + k);
        const v2f a1 = *(const v2f*)(aBase + (size_t)16 * HID + k);
        const v2f a2 = *(const v2f*)(aBase + (size_t)32 * HID + k);
        const v2f a3 = *(const v2f*)(aBase + (size_t)48 * HID + k);
        c0v = __builtin_amdgcn_wmma_f32_16x16x4_f32(false, a0, false, b,
                                                    (short)0, c0v, false, false);
        c1v = __builtin_amdgcn_wmma_f32_16x16x4_f32(false, a1, false, b,
                                                    (short)0, c1v, false, false);
        c2v = __builtin_amdgcn_wmma_f32_16x16x4_f32(false, a2, false, b,
                                                    (short)0, c2v, false, false);
        c3v = __builtin_amdgcn_wmma_f32_16x16x4_f32(false, a3, false, b,
                                                    (short)0, c3v, false, false);
    }

    const float bo = b_out[n0 + mn];
#pragma unroll
    for (int r = 0; r < 8; ++r) {
        const int m0 = t0 + r + hi * 8;
        out[(size_t)(m0     ) * VOCAB + n0 + mn] = c0v[r] + bo;
        out[(size_t)(m0 + 16) * VOCAB + n0 + mn] = c1v[r] + bo;
        out[(size_t)(m0 + 32) * VOCAB + n0 + mn] = c2v[r] + bo;
        out[(size_t)(m0 + 48) * VOCAB + n0 + mn] = c3v[r] + bo;
    }
}

// ---------------------------------------------------------------------------
// Launch. d_in order: inputs, W_ih, W_hh, b_ih, b_hh, W_out, b_out, h0, c0.
// d_ws layout: [0, 64MB) hs history, then 4-byte barrier counter.
// ---------------------------------------------------------------------------
extern "C" void kernel_launch(void* const* d_in, const int* in_sizes, int n_in,
                              void* d_out, int out_size, void* d_ws, size_t ws_size,
                              hipStream_t stream) {
    const float* inputs = (const float*)d_in[0];
    const float* W_ih   = (const float*)d_in[1];
    const float* W_hh   = (const float*)d_in[2];
    const float* b_ih   = (const float*)d_in[3];
    const float* b_hh   = (const float*)d_in[4];
    const float* W_out  = (const float*)d_in[5];
    const float* b_out  = (const float*)d_in[6];
    const float* h0     = (const float*)d_in[7];
    const float* c0     = (const float*)d_in[8];
    float* out = (float*)d_out;

    float*    hs      = (float*)d_ws;
    unsigned* counter = (unsigned*)((char*)d_ws + (size_t)SEQ * HID * sizeof(float));

    // barrier counter must start at 0 on every (graph-replayed) call
    hipMemsetAsync(counter, 0, sizeof(unsigned), stream);

    lstm_scan<<<NWG, TPB, 0, stream>>>(inputs, W_ih, W_hh, b_ih, b_hh,
                                       h0, c0, hs, counter);
    lstm_out_gemm<<<SEQ / 64, 256, 0, stream>>>(hs, W_out, b_out, out);
}